// KernelAttention_42554535969311
// MI455X (gfx1250) — compile-verified
//
#include <hip/hip_runtime.h>
#include <math.h>

// ---- problem constants (match reference) ----
constexpr int kWIN  = 4;     // windows (2x2)
constexpr int kC    = 64;    // channels
constexpr int kHD   = 8;     // head dim
constexpr int kS    = 36;    // WIN*3*3 token sequence
constexpr int kB    = 8;     // batch
constexpr int kRES  = 256;   // image side
constexpr int kWSZ  = 128;   // window side
constexpr int kTC   = 192;   // 3*C
constexpr int kKW   = 9 * kC * kC;   // floats per window kernel = 36864

typedef __attribute__((ext_vector_type(2))) float v2f;
typedef __attribute__((ext_vector_type(4))) float v4f;
typedef __attribute__((ext_vector_type(8))) float v8f;

// =====================================================================
// Stage 1: attention over kernel tokens. One block per c_out (64 blocks).
// Writes pre-SE kernels to ws as kern0[w][tap][co][ci].
// =====================================================================
__global__ void __launch_bounds__(128)
ka_attn_kernel(const float* __restrict__ conv_w,
               const float* __restrict__ qkv_w,
               const float* __restrict__ qkv_b,
               const float* __restrict__ proj_w,
               const float* __restrict__ proj_b,
               float* __restrict__ kern0)
{
    __shared__ float s_tok[kS * kC];   // tokens (s, ci)
    __shared__ float s_qkv[kS * kTC];  // qkv    (s, 3C)
    __shared__ float s_o[kS * kC];     // merged attn out (s, c)
    const int co  = blockIdx.x;
    const int tid = threadIdx.x;
    const int NT  = 128;

    // gather kernel tokens: tok[s,ci] = conv_w[w, co, ci, t], s = w*9+t
    for (int i = tid; i < kS * kC; i += NT) {
        int s = i >> 6, ci = i & 63;
        int w = s / 9, t = s % 9;
        s_tok[i] = conv_w[(((w * kC + co) * kC + ci) * 9) + t];
    }
    __syncthreads();

    // qkv = tok @ qkv_w + qkv_b
    for (int i = tid; i < kS * kTC; i += NT) {
        int s = i / kTC, j = i % kTC;
        float acc = qkv_b[j];
        const float* tr = &s_tok[s * kC];
        for (int ci = 0; ci < kC; ++ci)
            acc = fmaf(tr[ci], qkv_w[ci * kTC + j], acc);
        s_qkv[i] = acc;
    }
    __syncthreads();

    // per-(head, row) softmax attention; hd = 8, scale = 8^-0.5
    const float scale = 0.35355339059327373f;
    for (int i = tid; i < 8 * kS; i += NT) {
        int h = i / kS, s = i % kS;
        int h8 = h * kHD;
        float q[kHD];
        #pragma unroll
        for (int d = 0; d < kHD; ++d) q[d] = s_qkv[s * kTC + h8 + d] * scale;
        float L[kS];
        float m = -3.4e38f;
        #pragma unroll
        for (int t = 0; t < kS; ++t) {
            float lt = 0.f;
            #pragma unroll
            for (int d = 0; d < kHD; ++d)
                lt = fmaf(q[d], s_qkv[t * kTC + kC + h8 + d], lt);
            L[t] = lt;
            m = fmaxf(m, lt);
        }
        float den = 0.f;
        float o[kHD];
        #pragma unroll
        for (int d = 0; d < kHD; ++d) o[d] = 0.f;
        #pragma unroll
        for (int t = 0; t < kS; ++t) {
            float p = __expf(L[t] - m);
            den += p;
            #pragma unroll
            for (int d = 0; d < kHD; ++d)
                o[d] = fmaf(p, s_qkv[t * kTC + 2 * kC + h8 + d], o[d]);
        }
        float inv = 1.f / den;
        #pragma unroll
        for (int d = 0; d < kHD; ++d) s_o[s * kC + h8 + d] = o[d] * inv;
    }
    __syncthreads();

    // projection, write to kern0[w][tap][co][ci]
    for (int i = tid; i < kS * kC; i += NT) {
        int s = i >> 6, j = i & 63;
        float acc = proj_b[j];
        const float* orow = &s_o[s * kC];
        for (int c = 0; c < kC; ++c)
            acc = fmaf(orow[c], proj_w[c * kC + j], acc);
        int w = s / 9, t = s % 9;
        kern0[((w * 9 + t) * kC + co) * kC + j] = acc;
    }
}

// =====================================================================
// Stage 2: SE scaling; emit scaled kernel in WMMA *fragment order*:
//   kernF[w][((tap*4+mt)*32+lane)*32 + k4*2 + q]
//     = kern0[w][tap][co=mt*16+(lane&15)][ci=k4*4+2*(lane>>4)+q] * s(co,ci)
// so stage 3 can fetch A fragments with ds_load_b128 (2 K-steps/load).
// grid = WIN, block = 64 (thread == co).
// =====================================================================
__global__ void __launch_bounds__(64)
ka_se_kernel(const float* __restrict__ se_w1,
             const float* __restrict__ se_b1,
             const float* __restrict__ se_w2,
             const float* __restrict__ se_b2,
             const float* __restrict__ kern0,
             float* __restrict__ kernF)
{
    const int w  = blockIdx.x;
    const int co = threadIdx.x;
    const int mt = co >> 4, lanelo = co & 15;

    float h[4];
    #pragma unroll
    for (int d = 0; d < 4; ++d) h[d] = se_b1[w * 4 + d];
    for (int ci = 0; ci < kC; ++ci) {
        float p = 0.f;
        #pragma unroll
        for (int t = 0; t < 9; ++t)
            p += kern0[((w * 9 + t) * kC + co) * kC + ci];
        p *= (1.f / 9.f);
        #pragma unroll
        for (int d = 0; d < 4; ++d)
            h[d] = fmaf(p, se_w1[(w * 4 + d) * kC + ci], h[d]);
    }
    #pragma unroll
    for (int d = 0; d < 4; ++d) h[d] = fmaxf(h[d], 0.f);

    for (int ci = 0; ci < kC; ++ci) {
        float sv = se_b2[w * kC + ci];
        #pragma unroll
        for (int d = 0; d < 4; ++d)
            sv = fmaf(h[d], se_w2[(w * kC + ci) * 4 + d], sv);
        float sig = 1.f / (1.f + __expf(-sv));
        const int lane = lanelo + (((ci >> 1) & 1) << 4);
        const int k4   = ci >> 2, q = ci & 1;
        #pragma unroll
        for (int t = 0; t < 9; ++t)
            kernF[(size_t)w * kKW + ((t * 4 + mt) * 32 + lane) * 32 + k4 * 2 + q]
                = kern0[((w * 9 + t) * kC + co) * kC + ci] * sig;
    }
}

// ---------------------------------------------------------------------
// Async memory->LDS bulk copy (GLOBAL_LOAD_ASYNC_TO_LDS_B128, GVS mode).
// The same signed immediate applies to both LDS and global addresses.
// ---------------------------------------------------------------------
template <int I, int N>
__device__ __forceinline__ void async_fill_b128(unsigned l0, unsigned g0,
                                                const float* gbase)
{
    asm volatile("global_load_async_to_lds_b128 %0, %1, %2 offset:%3"
                 :: "v"(l0), "v"(g0), "s"(gbase), "n"(I * 4096)
                 : "memory");
    if constexpr (I + 1 < N) async_fill_b128<I + 1, N>(l0, g0, gbase);
}

// =====================================================================
// Stage 3: dynamic 3x3 conv per window via fp32 WMMA (9 shifted GEMMs).
// grid = (8 row-groups, B, WIN), block = 256 (8 waves).
// Wave tile: M=64 (4 x 16 co) x N=16 pixels; K = 9 taps x 64 ci.
// Window kernel (147 KB, fragment order) staged in LDS via async loads.
// Padding handled branchlessly: clamped column pointers + cndmask zeroing.
// =====================================================================
__global__ void __launch_bounds__(256)
ka_conv_kernel(const float* __restrict__ x,
               const float* __restrict__ kernF,
               float* __restrict__ y)
{
    extern __shared__ float sk[];          // 36864 floats = 147456 B
    const int tg   = blockIdx.x;           // row group (16 rows)
    const int b    = blockIdx.y;
    const int w    = blockIdx.z;
    const int wy   = w >> 1, wx = w & 1;
    const int tid  = threadIdx.x;
    const int wv   = tid >> 5;             // wave id 0..7
    const int lane = tid & 31;
    const int n    = lane & 15;            // fragment column / co row
    const int hi   = lane >> 4;            // K-half selector

    // async stage of this window's kernel into LDS (36 x b128 per lane)
    {
        unsigned l0 = (unsigned)(unsigned long long)sk + (unsigned)tid * 16u;
        unsigned g0 = (unsigned)tid * 16u;
        const float* gbase = kernF + (size_t)w * kKW;
        async_fill_b128<0, 36>(l0, g0, gbase);
        asm volatile("s_wait_asynccnt 0x0" ::: "memory");
    }
    __syncthreads();

    const size_t img  = (size_t)b * (kRES * kRES);
    const int    row0 = wy * kWSZ, col0 = wx * kWSZ;
    const v8f    vzero = {0.f, 0.f, 0.f, 0.f, 0.f, 0.f, 0.f, 0.f};
    const float* skl = sk + lane * 32;     // per-lane A-fragment base

    #pragma unroll 1
    for (int rr = 0; rr < 2; ++rr) {
        const int r = tg * 16 + wv * 2 + rr;          // output row in window
        #pragma unroll 1
        for (int st = 0; st < 8; ++st) {
            const int c0  = st * 16;                  // output column strip
            const int cin = c0 + n;                   // this lane's pixel col
            v8f acc[4];
            #pragma unroll
            for (int mt = 0; mt < 4; ++mt) acc[mt] = vzero;

            // clamped per-lane column pointers (always in-window)
            const bool okL = cin >= 1;
            const bool okR = cin <= kWSZ - 2;
            const float* pC = x + (img + (size_t)(row0 + r) * kRES
                                       + (size_t)(col0 + cin)) * kC + 2 * hi;
            const float* pL = okL ? (pC - kC) : pC;
            const float* pR = okR ? (pC + kC) : pC;

            #pragma unroll 1
            for (int dyi = 0; dyi < 3; ++dyi) {       // kernel row (dy = dyi-1)
                const int rin = r + dyi - 1;
                if (rin < 0 || rin >= kWSZ) continue; // uniform scalar branch
                const int roff = (dyi - 1) * (kRES * kC);
                const float* qL = pL + roff;
                const float* qC = pC + roff;
                const float* qR = pR + roff;
                const float* ska = skl + dyi * 3 * 4096;  // tap-row A base

                #pragma unroll 1
                for (int j = 0; j < 8; ++j) {         // two K-steps per iter
                    // B fragments for the 3 taps of this kernel row
                    v2f bL0 = *(const v2f*)(qL + j * 8);
                    v2f bL1 = *(const v2f*)(qL + j * 8 + 4);
                    v2f bC0 = *(const v2f*)(qC + j * 8);
                    v2f bC1 = *(const v2f*)(qC + j * 8 + 4);
                    v2f bR0 = *(const v2f*)(qR + j * 8);
                    v2f bR1 = *(const v2f*)(qR + j * 8 + 4);
                    // branchless window zero-pad (cndmask)
                    bL0.x = okL ? bL0.x : 0.f;  bL0.y = okL ? bL0.y : 0.f;
                    bL1.x = okL ? bL1.x : 0.f;  bL1.y = okL ? bL1.y : 0.f;
                    bR0.x = okR ? bR0.x : 0.f;  bR0.y = okR ? bR0.y : 0.f;
                    bR1.x = okR ? bR1.x : 0.f;  bR1.y = okR ? bR1.y : 0.f;

                    #pragma unroll
                    for (int mt = 0; mt < 4; ++mt) {
                        // A fragments: ds_load_b128 = 2 K-steps each
                        const v4f aL = *(const v4f*)(ska + (0 * 4 + mt) * 1024 + j * 4);
                        const v4f aC = *(const v4f*)(ska + (1 * 4 + mt) * 1024 + j * 4);
                        const v4f aR = *(const v4f*)(ska + (2 * 4 + mt) * 1024 + j * 4);
                        const v2f aL0 = {aL.x, aL.y}, aL1 = {aL.z, aL.w};
                        const v2f aC0 = {aC.x, aC.y}, aC1 = {aC.z, aC.w};
                        const v2f aR0 = {aR.x, aR.y}, aR1 = {aR.z, aR.w};
                        acc[mt] = __builtin_amdgcn_wmma_f32_16x16x4_f32(
                            false, aL0, false, bL0, (short)0, acc[mt], false, false);
                        acc[mt] = __builtin_amdgcn_wmma_f32_16x16x4_f32(
                            false, aL1, false, bL1, (short)0, acc[mt], false, false);
                        acc[mt] = __builtin_amdgcn_wmma_f32_16x16x4_f32(
                            false, aC0, false, bC0, (short)0, acc[mt], false, false);
                        acc[mt] = __builtin_amdgcn_wmma_f32_16x16x4_f32(
                            false, aC1, false, bC1, (short)0, acc[mt], false, false);
                        acc[mt] = __builtin_amdgcn_wmma_f32_16x16x4_f32(
                            false, aR0, false, bR0, (short)0, acc[mt], false, false);
                        acc[mt] = __builtin_amdgcn_wmma_f32_16x16x4_f32(
                            false, aR1, false, bR1, (short)0, acc[mt], false, false);
                    }
                }
            }

            // store: each lane holds 8 consecutive co for pixel n -> 2x b128
            float* yp = y + (img + (size_t)(row0 + r) * kRES
                                 + (size_t)(col0 + c0 + n)) * kC + hi * 8;
            #pragma unroll
            for (int mt = 0; mt < 4; ++mt) {
                const v4f* a4 = (const v4f*)&acc[mt];
                *(v4f*)(yp + mt * 16)     = a4[0];
                *(v4f*)(yp + mt * 16 + 4) = a4[1];
            }
        }
    }
}

// =====================================================================
extern "C" void kernel_launch(void* const* d_in, const int* in_sizes, int n_in,
                              void* d_out, int out_size, void* d_ws, size_t ws_size,
                              hipStream_t stream) {
    const float* x      = (const float*)d_in[0];
    const float* conv_w = (const float*)d_in[1];
    const float* qkv_w  = (const float*)d_in[2];
    const float* qkv_b  = (const float*)d_in[3];
    const float* proj_w = (const float*)d_in[4];
    const float* proj_b = (const float*)d_in[5];
    const float* se_w1  = (const float*)d_in[6];
    const float* se_b1  = (const float*)d_in[7];
    const float* se_w2  = (const float*)d_in[8];
    const float* se_b2  = (const float*)d_in[9];
    float* y     = (float*)d_out;
    float* kern0 = (float*)d_ws;                     // raw scaled-input kernels
    float* kernF = kern0 + (size_t)kWIN * kKW;       // fragment-order kernels

    // Stage 1: kernel-token attention (64 independent c_out batches)
    ka_attn_kernel<<<kC, 128, 0, stream>>>(conv_w, qkv_w, qkv_b,
                                           proj_w, proj_b, kern0);
    // Stage 2: SE scaling + fragment-order repack
    ka_se_kernel<<<kWIN, kC, 0, stream>>>(se_w1, se_b1, se_w2, se_b2,
                                          kern0, kernF);

    // Stage 3: WMMA dynamic conv
    const size_t shmem = (size_t)kKW * sizeof(float); // 147456 B
    (void)hipFuncSetAttribute(reinterpret_cast<const void*>(ka_conv_kernel),
                              hipFuncAttributeMaxDynamicSharedMemorySize,
                              (int)shmem);
    ka_conv_kernel<<<dim3(8, kB, kWIN), 256, shmem, stream>>>(x, kernF, y);
}